// NetVLAD_46557445489054
// MI455X (gfx1250) — compile-verified
//
#include <hip/hip_runtime.h>
#include <hip/hip_bf16.h>
#include <stdint.h>

typedef __attribute__((ext_vector_type(2))) float v2f;
typedef __attribute__((ext_vector_type(8))) float v8f;

#define B_SZ   32
#define C_SZ   128
#define K_SZ   64
#define N_SZ   16384
#define SPLITS 32
#define NT     (N_SZ / SPLITS)   // 512 columns per block
#define TILE_N 64                // columns per block-iteration
#define NITER  (NT / TILE_N)     // 8

#define SW_STRIDE 129            // W rows padded: bank = k*129+c -> distinct over k
#define SX_STRIDE 68             // x rows: 272B (16B-aligned for async B128);
                                 //   GEMM2 banks = (4c+nk)%64 distinct over 16 lanes
#define SA_STRIDE 65

__device__ __forceinline__ v8f wmma_f32(v2f a, v2f b, v8f c) {
    // D(16x16,f32) = A(16x4,f32) * B(4x16,f32) + C
    return __builtin_amdgcn_wmma_f32_16x16x4_f32(
        /*neg_a=*/false, a, /*neg_b=*/false, b,
        /*c_mod=*/(short)0, c, /*reuse_a=*/false, /*reuse_b=*/false);
}

// ASYNCcnt-tracked DMA: 16B global -> LDS, no VGPR round-trip (ISA §15.18 op 98)
__device__ __forceinline__ void async_ldsload_b128(uint32_t lds_off, const float* g) {
    asm volatile("global_load_async_to_lds_b128 %0, %1, off"
                 :: "v"(lds_off), "v"((const void*)g) : "memory");
}
__device__ __forceinline__ void wait_async_le8() {
    asm volatile("s_wait_asynccnt 0x8" ::: "memory");
}
__device__ __forceinline__ void wait_async_0() {
    asm volatile("s_wait_asynccnt 0x0" ::: "memory");
}

__global__ __launch_bounds__(256, 2)
void netvlad_stage1(const float* __restrict__ x,
                    const float* __restrict__ conv_w,
                    const float* __restrict__ conv_b,
                    float* __restrict__ vlad_ws,   // (B, K, C) f32 partial
                    float* __restrict__ asum_ws)   // (B, K)    f32 partial
{
    __shared__ float sW[K_SZ * SW_STRIDE];           // 64 x 128 (padded)
    __shared__ float sX[2][C_SZ * SX_STRIDE];        // double-buffered 128 x 64
    __shared__ float sA[K_SZ * SA_STRIDE];           // 64 x 64 logits -> softmax
    __shared__ float sBias[K_SZ];
    __shared__ float sAsum[K_SZ];

    const int tid   = threadIdx.x;
    const int wave  = tid >> 5;          // 0..7
    const int lane  = tid & 31;
    const int lan16 = lane & 15;
    const int half2 = (lane >> 4) * 2;   // K offset inside a 4-step (0 or 2)

    const int b     = blockIdx.x >> 5;
    const int split = blockIdx.x & 31;
    const long long n_base = (long long)split * NT;
    const float* __restrict__ xb = x + (long long)b * C_SZ * N_SZ;

    // ---- load W (64x128) and bias into LDS, zero asum ----
    for (int i = tid; i < K_SZ * C_SZ; i += 256) {
        int k = i >> 7, c = i & 127;
        sW[k * SW_STRIDE + c] = conv_w[k * C_SZ + c];
    }
    if (tid < K_SZ) { sBias[tid] = conv_b[tid]; sAsum[tid] = 0.0f; }

    // GEMM2 ownership: wave w -> k-tile (w>>1), c-tiles 4*(w&1)..+3
    const int kt2     = wave >> 1;
    const int ct_base = (wave & 1) * 4;
    v8f vacc[4] = {};                    // 4 x (16x16) f32 accumulators

    // GEMM1 ownership: wave w -> k-tile (w>>1), n-tiles 2*(w&1)..+1
    const int kt1 = wave >> 1;
    const int ntA = (wave & 1) * 2;

    // ---- async DMA stage of one 128x64 x-tile: 8 B128 transfers per thread ----
    auto stage_async = [&](int it, int bufi) {
        const long long n0 = n_base + (long long)it * TILE_N;
        #pragma unroll
        for (int r = 0; r < 8; ++r) {
            const int i  = tid + r * 256;        // 2048 float4 chunks total
            const int c  = i >> 4;               // 16 float4 per row
            const int f4 = i & 15;
            const uint32_t lds_off =
                (uint32_t)(uintptr_t)&sX[bufi][c * SX_STRIDE + f4 * 4];
            async_ldsload_b128(lds_off, xb + (long long)c * N_SZ + n0 + f4 * 4);
        }
    };

    stage_async(0, 0);                    // prologue: tile 0 in flight

    for (int it = 0; it < NITER; ++it) {
        // issue next tile into the other buffer (its readers finished at the
        // barrier that ended the previous iteration), then drain current tile
        if (it + 1 < NITER) {
            stage_async(it + 1, (it + 1) & 1);
            wait_async_le8();             // in-order: leaves only the 8 new ones
        } else {
            wait_async_0();
        }
        __syncthreads();                  // make all waves' DMA visible block-wide

        const float* __restrict__ sXc = sX[it & 1];

        // ---- GEMM1: logits(64x64) = W(64x128) @ xt(128x64), f32 WMMA ----
        {
            v8f l0 = {}, l1 = {};
            const int rowA = kt1 * 16 + lan16;     // A: M = lane%16
            const int nc0  = ntA * 16 + lan16;     // B: N = lane%16
            #pragma unroll
            for (int s = 0; s < 32; ++s) {
                const int c0 = 4 * s + half2;      // reduction index
                v2f a;
                a.x = sW[rowA * SW_STRIDE + c0];
                a.y = sW[rowA * SW_STRIDE + c0 + 1];
                v2f b0;
                b0.x = sXc[c0 * SX_STRIDE + nc0];
                b0.y = sXc[(c0 + 1) * SX_STRIDE + nc0];
                l0 = wmma_f32(a, b0, l0);
                v2f b1;
                b1.x = sXc[c0 * SX_STRIDE + nc0 + 16];
                b1.y = sXc[(c0 + 1) * SX_STRIDE + nc0 + 16];
                l1 = wmma_f32(a, b1, l1);
            }
            // write logits + bias to sA  (C/D: M = v + 8*(lane/16), N = lane%16)
            #pragma unroll
            for (int v = 0; v < 8; ++v) {
                const int row = kt1 * 16 + v + 8 * (lane >> 4);
                const float bias = sBias[row];
                sA[row * SA_STRIDE + ntA * 16 + lan16]       = l0[v] + bias;
                sA[row * SA_STRIDE + (ntA + 1) * 16 + lan16] = l1[v] + bias;
            }
        }
        __syncthreads();

        // ---- softmax(relu(.)) over K per column; one thread per column ----
        if (tid < TILE_N) {
            const int n = tid;
            float m = 0.0f;                        // relu() >= 0
            #pragma unroll 8
            for (int k = 0; k < K_SZ; ++k) {
                float r = sA[k * SA_STRIDE + n];
                r = r > 0.0f ? r : 0.0f;
                m = fmaxf(m, r);
            }
            float sum = 0.0f;
            #pragma unroll 8
            for (int k = 0; k < K_SZ; ++k) {
                float r = sA[k * SA_STRIDE + n];
                r = r > 0.0f ? r : 0.0f;
                float e = __expf(r - m);
                sA[k * SA_STRIDE + n] = e;
                sum += e;
            }
            const float inv = 1.0f / sum;
            #pragma unroll 8
            for (int k = 0; k < K_SZ; ++k) sA[k * SA_STRIDE + n] *= inv;
        }
        __syncthreads();

        // ---- asum accumulation: thread k sums row k of A ----
        if (tid < K_SZ) {
            float s = 0.0f;
            #pragma unroll 8
            for (int n = 0; n < TILE_N; ++n) s += sA[tid * SA_STRIDE + n];
            sAsum[tid] += s;
        }

        // ---- GEMM2: vlad(64x128) += A(64x64) @ xtT(64x128), f32 WMMA ----
        {
            const int rowA2 = kt2 * 16 + lan16;    // A: M = lane%16 (k-dim)
            #pragma unroll
            for (int s = 0; s < 16; ++s) {
                const int nk = 4 * s + half2;      // reduction index (n in tile)
                v2f a;
                a.x = sA[rowA2 * SA_STRIDE + nk];
                a.y = sA[rowA2 * SA_STRIDE + nk + 1];
                #pragma unroll
                for (int t = 0; t < 4; ++t) {
                    const int ccol = (ct_base + t) * 16 + lan16;  // B: N = c
                    v2f bb;
                    bb.x = sXc[ccol * SX_STRIDE + nk];    // X^T[nk][c] = sX[c][nk]
                    bb.y = sXc[ccol * SX_STRIDE + nk + 1];
                    vacc[t] = wmma_f32(a, bb, vacc[t]);
                }
            }
        }
        __syncthreads();   // readers of sXc / sA done before buffer reuse
    }

    // ---- flush partials with fp32 global atomics ----
    if (tid < K_SZ)
        unsafeAtomicAdd(&asum_ws[b * K_SZ + tid], sAsum[tid]);

    float* vbase = vlad_ws + (long long)b * K_SZ * C_SZ;
    #pragma unroll
    for (int t = 0; t < 4; ++t) {
        #pragma unroll
        for (int v = 0; v < 8; ++v) {
            const int row = kt2 * 16 + v + 8 * (lane >> 4);
            const int col = (ct_base + t) * 16 + lan16;
            unsafeAtomicAdd(&vbase[row * C_SZ + col], vacc[t][v]);
        }
    }
}

__global__ __launch_bounds__(256)
void netvlad_stage2(const float* __restrict__ vlad_ws,
                    const float* __restrict__ asum_ws,
                    const float* __restrict__ centroids,
                    float* __restrict__ out)
{
    const int b    = blockIdx.x;
    const int wave = threadIdx.x >> 5;
    const int lane = threadIdx.x & 31;
    __shared__ float s_gsum[8];

    const float* vb = vlad_ws + (long long)b * K_SZ * C_SZ;
    const float* as = asum_ws + b * K_SZ;
    float*       ob = out     + (long long)b * K_SZ * C_SZ;

    float vals[8][4];
    float gss = 0.0f;

    #pragma unroll
    for (int r = 0; r < 8; ++r) {
        const int k = wave * 8 + r;          // each wave owns 8 rows of K
        const float a = as[k];
        float ss = 0.0f;
        float tmp[4];
        #pragma unroll
        for (int j = 0; j < 4; ++j) {        // 128 cols / 32 lanes = 4 each
            const int c = lane * 4 + j;
            const float v = vb[k * C_SZ + c] - a * centroids[k * C_SZ + c];
            tmp[j] = v;
            ss += v * v;
        }
        #pragma unroll
        for (int off = 16; off > 0; off >>= 1) ss += __shfl_xor(ss, off, 32);
        const float inv = 1.0f / fmaxf(sqrtf(ss), 1e-12f);   // intra-normalize
        #pragma unroll
        for (int j = 0; j < 4; ++j) {
            tmp[j] *= inv;
            vals[r][j] = tmp[j];
            gss += tmp[j] * tmp[j];
        }
    }

    #pragma unroll
    for (int off = 16; off > 0; off >>= 1) gss += __shfl_xor(gss, off, 32);
    if (lane == 0) s_gsum[wave] = gss;
    __syncthreads();
    float g = 0.0f;
    #pragma unroll
    for (int w = 0; w < 8; ++w) g += s_gsum[w];
    const float ginv = 1.0f / fmaxf(sqrtf(g), 1e-12f);       // global normalize

    #pragma unroll
    for (int r = 0; r < 8; ++r) {
        const int k = wave * 8 + r;
        #pragma unroll
        for (int j = 0; j < 4; ++j)
            ob[k * C_SZ + lane * 4 + j] = vals[r][j] * ginv;
    }
}

extern "C" void kernel_launch(void* const* d_in, const int* in_sizes, int n_in,
                              void* d_out, int out_size, void* d_ws, size_t ws_size,
                              hipStream_t stream) {
    const float* x         = (const float*)d_in[0];   // (32,128,16384,1)
    const float* conv_w    = (const float*)d_in[1];   // (64,128)
    const float* conv_b    = (const float*)d_in[2];   // (64,)
    const float* centroids = (const float*)d_in[3];   // (64,128)
    float* out = (float*)d_out;                       // (32, 8192)

    float* vlad_ws = (float*)d_ws;                    // B*K*C f32 = 1 MB
    float* asum_ws = vlad_ws + (size_t)B_SZ * K_SZ * C_SZ;
    const size_t zero_bytes =
        ((size_t)B_SZ * K_SZ * C_SZ + (size_t)B_SZ * K_SZ) * sizeof(float);
    hipMemsetAsync(d_ws, 0, zero_bytes, stream);

    netvlad_stage1<<<dim3(B_SZ * SPLITS), dim3(256), 0, stream>>>(
        x, conv_w, conv_b, vlad_ws, asum_ws);
    netvlad_stage2<<<dim3(B_SZ), dim3(256), 0, stream>>>(
        vlad_ws, asum_ws, centroids, out);
}